// DenoisingEdgeNetwork_83700322665115
// MI455X (gfx1250) — compile-verified
//
#include <hip/hip_runtime.h>
#include <hip/hip_bf16.h>

#define NN    2048
#define EE    262144
#define SDIM  256
#define VDIM  64
#define EDIM  32
#define NAF   16
#define NBT   5
#define NMOL  64

typedef float v2f __attribute__((ext_vector_type(2)));
typedef float v8f __attribute__((ext_vector_type(8)));

// fast SiLU: v_exp + v_add + v_rcp + v_mul (avoid IEEE div expansion)
__device__ __forceinline__ float silu_f(float x) {
    float den = 1.0f + __expf(-x);
    return x * __builtin_amdgcn_rcpf(den);
}

// ---------------------------------------------------------------------------
// s_h = silu(s @ Ws + bs)   [N,256] = [N,256]@[256,256]
// one wave per 16x16 output tile, fp32 WMMA 16x16x4, K-loop of 64
// ---------------------------------------------------------------------------
__global__ void __launch_bounds__(256) k_sh(const float* __restrict__ s,
                                            const float* __restrict__ Ws,
                                            const float* __restrict__ bs,
                                            float* __restrict__ sh) {
    int wave  = (blockIdx.x * blockDim.x + threadIdx.x) >> 5;   // 0..2047
    int lane  = threadIdx.x & 31;
    int half  = lane >> 4;
    int lm    = lane & 15;
    int mtile = wave >> 4;        // 0..127
    int ntile = wave & 15;        // 0..15
    int m0 = mtile * 16, n0 = ntile * 16;

    float bias = bs[n0 + lm];
    v8f acc;
#pragma unroll
    for (int r = 0; r < 8; ++r) acc[r] = bias;

    const float* arow = s + (m0 + lm) * SDIM;
    for (int kk = 0; kk < SDIM / 4; ++kk) {
        int k = kk * 4 + 2 * half;
        v2f a = *(const v2f*)(arow + k);
        v2f b;
        b.x = Ws[k * SDIM + n0 + lm];
        b.y = Ws[(k + 1) * SDIM + n0 + lm];
        acc = __builtin_amdgcn_wmma_f32_16x16x4_f32(false, a, false, b,
                                                    (short)0, acc, false, false);
    }
#pragma unroll
    for (int r = 0; r < 8; ++r) {
        int M = r + 8 * half;
        sh[(m0 + M) * SDIM + n0 + lm] = silu_f(acc[r]);
    }
}

// ---------------------------------------------------------------------------
// atoms_pred = s_h @ Wa + ba   [N,16] ; one wave per 16-row tile (128 waves)
// ---------------------------------------------------------------------------
__global__ void __launch_bounds__(256) k_atoms(const float* __restrict__ sh,
                                               const float* __restrict__ Wa,
                                               const float* __restrict__ ba,
                                               float* __restrict__ out_atoms) {
    int wave = (blockIdx.x * blockDim.x + threadIdx.x) >> 5;    // 0..127
    int lane = threadIdx.x & 31;
    int half = lane >> 4;
    int lm   = lane & 15;
    int m0 = wave * 16;

    float bias = ba[lm];
    v8f acc;
#pragma unroll
    for (int r = 0; r < 8; ++r) acc[r] = bias;

    const float* arow = sh + (m0 + lm) * SDIM;
    for (int kk = 0; kk < SDIM / 4; ++kk) {
        int k = kk * 4 + 2 * half;
        v2f a = *(const v2f*)(arow + k);
        v2f b;
        b.x = Wa[k * NAF + lm];
        b.y = Wa[(k + 1) * NAF + lm];
        acc = __builtin_amdgcn_wmma_f32_16x16x4_f32(false, a, false, b,
                                                    (short)0, acc, false, false);
    }
#pragma unroll
    for (int r = 0; r < 8; ++r) {
        int M = r + 8 * half;
        out_atoms[(m0 + M) * NAF + lm] = acc[r];
    }
}

// ---------------------------------------------------------------------------
// coords_pred_raw = p + (v @ Wc)
// ---------------------------------------------------------------------------
__global__ void k_coords(const float* __restrict__ v, const float* __restrict__ Wc,
                         const float* __restrict__ p, float* __restrict__ cp) {
    int t = blockIdx.x * blockDim.x + threadIdx.x;
    if (t >= NN * 3) return;
    int n = t / 3, ax = t % 3;
    const float* vr = v + (n * 3 + ax) * VDIM;
    float acc = 0.0f;
#pragma unroll 8
    for (int k = 0; k < VDIM; ++k) acc += vr[k] * Wc[k];
    cp[t] = p[t] + acc;
}

// per-molecule mean (deterministic; 64 threads, one per molecule)
__global__ void k_mean(const float* __restrict__ cp, const int* __restrict__ batch,
                       float* __restrict__ mean) {
    int m = threadIdx.x;   // 0..63
    float sx = 0, sy = 0, sz = 0, cnt = 0;
    for (int n = 0; n < NN; ++n) {
        if (batch[n] == m) {
            sx += cp[n * 3 + 0];
            sy += cp[n * 3 + 1];
            sz += cp[n * 3 + 2];
            cnt += 1.0f;
        }
    }
    float inv = 1.0f / fmaxf(cnt, 1.0f);
    mean[m * 3 + 0] = sx * inv;
    mean[m * 3 + 1] = sy * inv;
    mean[m * 3 + 2] = sz * inv;
}

__global__ void k_cpfinal(float* __restrict__ cp, const int* __restrict__ batch,
                          const float* __restrict__ mean, float* __restrict__ out_coords) {
    int t = blockIdx.x * blockDim.x + threadIdx.x;
    if (t >= NN * 3) return;
    int n = t / 3, ax = t % 3;
    float val = cp[t] - mean[batch[n] * 3 + ax];
    cp[t] = val;
    out_coords[t] = val;
}

// ---------------------------------------------------------------------------
// edge-id map for dense symmetrization (no 536MB tensor):
//   idxmat[a*N+b] = max edge k with (j_k,i_k)==(a,b), else -1
// ---------------------------------------------------------------------------
__global__ void k_idxinit(int* __restrict__ idxmat) {
    int t = blockIdx.x * blockDim.x + threadIdx.x;
    if (t < NN * NN) idxmat[t] = -1;
}

__global__ void k_scatter(const int* __restrict__ eig, int* __restrict__ idxmat) {
    int k = blockIdx.x * blockDim.x + threadIdx.x;
    if (k >= EE) return;
    int j = eig[k];
    int i = eig[EE + k];
    atomicMax(&idxmat[j * NN + i], k);
}

// ---------------------------------------------------------------------------
// Fused edge kernel: per wave = 16 edges.
//   e_sym -> f = s_h[i]+s_h[j]+e_sym@Wb+bb  (WMMA, K=32)
//   hidden = silu([f|d] @ W0 + b0)          (WMMA, K=256 + rank-1 d update)
//   bonds  = hidden @ W1 + b1               (LDS + VALU, 16x5)
// Block = 64 threads (2 waves), ~61.5KB static LDS.
// ---------------------------------------------------------------------------
__global__ void __launch_bounds__(64) k_edges(
        const float* __restrict__ sh, const float* __restrict__ cp,
        const float* __restrict__ e,  const int* __restrict__ eig,
        const int* __restrict__ idxmat,
        const float* __restrict__ Wb, const float* __restrict__ bb,
        const float* __restrict__ W0, const float* __restrict__ b0,
        const float* __restrict__ W1, const float* __restrict__ b1,
        float* __restrict__ out_bonds) {
    __shared__ float W0blk[SDIM * 16];      // 16KB: W0[:, ntile*16 : +16]
    __shared__ float w0last[SDIM];          // 1KB : W0[256, :]
    __shared__ float b0l[SDIM];             // 1KB
    __shared__ float W1l[SDIM * NBT];       // 5KB
    __shared__ float b1l[NBT];
    __shared__ float f_lds[2][16 * SDIM];   // 32KB
    __shared__ float esym[2][16 * EDIM];    // 4KB
    __shared__ float hid[2][16 * 16];       // 2KB
    __shared__ float dl[2][16];
    __shared__ int   il[2][16], jl[2][16];

    int w    = threadIdx.x >> 5;            // wave in block (0/1)
    int lane = threadIdx.x & 31;
    int half = lane >> 4;
    int lm   = lane & 15;
    int ebase = blockIdx.x * 32 + w * 16;   // first edge of this wave's tile

    // ---- Phase 1a: cooperative parameter staging + per-wave edge meta ----
    for (int idx = threadIdx.x; idx < SDIM; idx += 64) {
        b0l[idx]    = b0[idx];
        w0last[idx] = W0[SDIM * SDIM + idx];    // row 256 of W0 [257,256]
    }
    for (int idx = threadIdx.x; idx < SDIM * NBT; idx += 64) W1l[idx] = W1[idx];
    if (threadIdx.x < NBT) b1l[threadIdx.x] = b1[threadIdx.x];

    if (lane < 16) {
        int k  = ebase + lane;
        int jj = eig[k];
        int ii = eig[EE + k];
        jl[w][lane] = jj;
        il[w][lane] = ii;
        float dx = cp[ii * 3 + 0] - cp[jj * 3 + 0];
        float dy = cp[ii * 3 + 1] - cp[jj * 3 + 1];
        float dz = cp[ii * 3 + 2] - cp[jj * 3 + 2];
        dl[w][lane] = dx * dx + dy * dy + dz * dz;
    }
    __syncthreads();

    // ---- Phase 1b: e_sym tile (16x32) and f init g = sh[i]+sh[j]+bb ----
    {
        int row = lm;
        int ii = il[w][row], jj = jl[w][row];
        int m1 = idxmat[jj * NN + ii];
        int m2 = idxmat[ii * NN + jj];
        const float* e1 = e + (m1 >= 0 ? m1 : 0) * EDIM;
        const float* e2 = e + (m2 >= 0 ? m2 : 0) * EDIM;
        float f1 = (m1 >= 0) ? 0.5f : 0.0f;
        float f2 = (m2 >= 0) ? 0.5f : 0.0f;
#pragma unroll
        for (int c = 0; c < 16; ++c) {
            int col = half * 16 + c;
            esym[w][row * EDIM + col] = f1 * e1[col] + f2 * e2[col];
        }
    }
    for (int slot = lane; slot < 16 * SDIM / 4; slot += 32) {
        int row = slot >> 6;                 // 64 float4 per 256-wide row
        int c4  = (slot & 63) * 4;
        int ii = il[w][row], jj = jl[w][row];
        float4 A = *(const float4*)(sh + ii * SDIM + c4);
        float4 B = *(const float4*)(sh + jj * SDIM + c4);
        float4 C = *(const float4*)(bb + c4);
        float4 R;
        R.x = A.x + B.x + C.x;
        R.y = A.y + B.y + C.y;
        R.z = A.z + B.z + C.z;
        R.w = A.w + B.w + C.w;
        *(float4*)(&f_lds[w][row * SDIM + c4]) = R;
    }
    __syncthreads();

    // ---- Phase 2: f += e_sym @ Wb  (WMMA, K=32) ----
    for (int nt = 0; nt < 16; ++nt) {
        int n0 = nt * 16;
        v8f acc;
#pragma unroll
        for (int r = 0; r < 8; ++r)
            acc[r] = f_lds[w][(r + 8 * half) * SDIM + n0 + lm];
#pragma unroll
        for (int kk = 0; kk < EDIM / 4; ++kk) {
            int k = kk * 4 + 2 * half;
            v2f a = *(const v2f*)(&esym[w][lm * EDIM + k]);
            v2f b;
            b.x = Wb[k * SDIM + n0 + lm];
            b.y = Wb[(k + 1) * SDIM + n0 + lm];
            acc = __builtin_amdgcn_wmma_f32_16x16x4_f32(false, a, false, b,
                                                        (short)0, acc, false, false);
        }
#pragma unroll
        for (int r = 0; r < 8; ++r)
            f_lds[w][(r + 8 * half) * SDIM + n0 + lm] = acc[r];
    }
    __syncthreads();

    // ---- Phase 3: hidden = silu([f|d]@W0+b0); bonds += hidden_tile @ W1 ----
    float bacc0 = 0.0f, bacc1 = 0.0f, bacc2 = 0.0f;
    for (int nt = 0; nt < 16; ++nt) {
        int n0 = nt * 16;
        // cooperatively stage W0 column block (rows 0..255)
        for (int idx = threadIdx.x; idx < SDIM * 16; idx += 64)
            W0blk[idx] = W0[(idx >> 4) * SDIM + n0 + (idx & 15)];
        __syncthreads();

        v8f acc;
#pragma unroll
        for (int r = 0; r < 8; ++r) {
            int M = r + 8 * half;
            // bias + rank-1 update from the concatenated distance column
            acc[r] = b0l[n0 + lm] + dl[w][M] * w0last[n0 + lm];
        }
        for (int kk = 0; kk < SDIM / 4; ++kk) {
            int k = kk * 4 + 2 * half;
            v2f a = *(const v2f*)(&f_lds[w][lm * SDIM + k]);
            v2f b;
            b.x = W0blk[k * 16 + lm];
            b.y = W0blk[(k + 1) * 16 + lm];
            acc = __builtin_amdgcn_wmma_f32_16x16x4_f32(false, a, false, b,
                                                        (short)0, acc, false, false);
        }
#pragma unroll
        for (int r = 0; r < 8; ++r)
            hid[w][(r + 8 * half) * 16 + lm] = silu_f(acc[r]);
        __syncthreads();

        // bonds partial: 80 (m,c) pairs over 32 lanes
#pragma unroll
        for (int pp = 0; pp < 3; ++pp) {
            int mc = lane + pp * 32;
            if (mc < 16 * NBT) {
                int m = mc / NBT, c = mc % NBT;
                float sacc = 0.0f;
#pragma unroll
                for (int nl = 0; nl < 16; ++nl)
                    sacc += hid[w][m * 16 + nl] * W1l[(n0 + nl) * NBT + c];
                if (pp == 0) bacc0 += sacc;
                else if (pp == 1) bacc1 += sacc;
                else bacc2 += sacc;
            }
        }
        __syncthreads();
    }

    // ---- write bonds ----
#pragma unroll
    for (int pp = 0; pp < 3; ++pp) {
        int mc = lane + pp * 32;
        if (mc < 16 * NBT) {
            int m = mc / NBT, c = mc % NBT;
            float val = (pp == 0 ? bacc0 : pp == 1 ? bacc1 : bacc2) + b1l[c];
            out_bonds[(ebase + m) * NBT + c] = val;
        }
    }
}

// ---------------------------------------------------------------------------
extern "C" void kernel_launch(void* const* d_in, const int* in_sizes, int n_in,
                              void* d_out, int out_size, void* d_ws, size_t ws_size,
                              hipStream_t stream) {
    (void)in_sizes; (void)n_in; (void)out_size; (void)ws_size;

    const float* s     = (const float*)d_in[0];
    const float* v     = (const float*)d_in[1];
    const float* p     = (const float*)d_in[2];
    const float* e     = (const float*)d_in[3];
    const int*   batch = (const int*)d_in[4];
    const int*   eig   = (const int*)d_in[5];
    const float* Ws = (const float*)d_in[6];
    const float* bs = (const float*)d_in[7];
    const float* Wc = (const float*)d_in[8];
    const float* Wa = (const float*)d_in[9];
    const float* ba = (const float*)d_in[10];
    const float* Wb = (const float*)d_in[11];
    const float* bb = (const float*)d_in[12];
    const float* W0 = (const float*)d_in[13];
    const float* b0 = (const float*)d_in[14];
    const float* W1 = (const float*)d_in[15];
    const float* b1 = (const float*)d_in[16];

    float* out        = (float*)d_out;
    float* out_coords = out;                       // N*3
    float* out_atoms  = out + NN * 3;              // N*16
    float* out_bonds  = out + NN * 3 + NN * NAF;   // E*5

    // workspace: s_h | cp | mean | idxmat  (~19MB)
    float* ws_f   = (float*)d_ws;
    float* sh     = ws_f;                               // N*SDIM
    float* cp     = sh + NN * SDIM;                     // N*3
    float* mean   = cp + NN * 3;                        // NMOL*3
    int*   idxmat = (int*)(mean + NMOL * 3);            // N*N

    // s_h = silu(s@Ws+bs): 2048 waves -> 256 blocks x 256
    hipLaunchKernelGGL(k_sh, dim3(256), dim3(256), 0, stream, s, Ws, bs, sh);
    // atoms_pred: 128 waves -> 16 blocks x 256
    hipLaunchKernelGGL(k_atoms, dim3(16), dim3(256), 0, stream, sh, Wa, ba, out_atoms);

    // coords pipeline
    hipLaunchKernelGGL(k_coords, dim3((NN * 3 + 255) / 256), dim3(256), 0, stream,
                       v, Wc, p, cp);
    hipLaunchKernelGGL(k_mean, dim3(1), dim3(NMOL), 0, stream, cp, batch, mean);
    hipLaunchKernelGGL(k_cpfinal, dim3((NN * 3 + 255) / 256), dim3(256), 0, stream,
                       cp, batch, mean, out_coords);

    // edge-id map
    hipLaunchKernelGGL(k_idxinit, dim3(NN * NN / 256), dim3(256), 0, stream, idxmat);
    hipLaunchKernelGGL(k_scatter, dim3(EE / 256), dim3(256), 0, stream, eig, idxmat);

    // fused edge kernel: E/32 blocks of 64 threads (2 waves x 16 edges)
    hipLaunchKernelGGL(k_edges, dim3(EE / 32), dim3(64), 0, stream,
                       sh, cp, e, eig, idxmat, Wb, bb, W0, b0, W1, b1, out_bonds);
}